// GridPooling_Module_6365141532776
// MI455X (gfx1250) — compile-verified
//
#include <hip/hip_runtime.h>
#include <hip/hip_bf16.h>
#include <limits.h>

typedef __attribute__((ext_vector_type(2))) float v2f;
typedef __attribute__((ext_vector_type(8))) float v8f;

#define INV_VOXEL 20.0f      // 1.0/0.05 rounds to exactly 20.0
#define SLOTS 8192           // LDS histogram bins (>= 20^3 = 8000 for this data)
#define NSLICES 64           // histogram blocks == partial slices
#define HIST_THREADS 1024

// hdr layout (ints)
enum { HDR_MIN0 = 0, HDR_MIN1, HDR_MIN2, HDR_MAX0, HDR_MAX1, HDR_MAX2,
       HDR_H, HDR_D1, HDR_D2, HDR_NU, HDR_FLAG, HDR_WORDS = 16 };

#if __has_builtin(__builtin_amdgcn_wmma_f32_16x16x4_f32)
#define USE_WMMA_REDUCE 1
#endif

__global__ void k_init(int* hdr) {
  int t = threadIdx.x;
  if (t < 3) hdr[t] = INT_MAX;
  else if (t < 6) hdr[t] = INT_MIN;
  else if (t < HDR_WORDS) hdr[t] = 0;
}

__global__ void k_minmax(const float* __restrict__ x, int n, int* __restrict__ hdr) {
  __shared__ int smn[3], smx[3];
  int tid = threadIdx.x;
  if (tid < 3) { smn[tid] = INT_MAX; smx[tid] = INT_MIN; }
  __syncthreads();
  int l0 = INT_MAX, l1 = INT_MAX, l2 = INT_MAX;
  int h0 = INT_MIN, h1 = INT_MIN, h2 = INT_MIN;
  int stride = gridDim.x * blockDim.x;
  for (int i = blockIdx.x * blockDim.x + tid; i < n; i += stride) {
    float px = x[3 * i], py = x[3 * i + 1], pz = x[3 * i + 2];
    int v0 = (int)floorf(px * INV_VOXEL);
    int v1 = (int)floorf(py * INV_VOXEL);
    int v2 = (int)floorf(pz * INV_VOXEL);
    l0 = min(l0, v0); h0 = max(h0, v0);
    l1 = min(l1, v1); h1 = max(h1, v1);
    l2 = min(l2, v2); h2 = max(h2, v2);
  }
  atomicMin(&smn[0], l0); atomicMin(&smn[1], l1); atomicMin(&smn[2], l2);
  atomicMax(&smx[0], h0); atomicMax(&smx[1], h1); atomicMax(&smx[2], h2);
  __syncthreads();
  if (tid == 0) {
    atomicMin(&hdr[HDR_MIN0], smn[0]); atomicMin(&hdr[HDR_MIN1], smn[1]); atomicMin(&hdr[HDR_MIN2], smn[2]);
    atomicMax(&hdr[HDR_MAX0], smx[0]); atomicMax(&hdr[HDR_MAX1], smx[1]); atomicMax(&hdr[HDR_MAX2], smx[2]);
  }
}

__global__ void k_dims(int* hdr, int cap) {
  if (threadIdx.x == 0 && blockIdx.x == 0) {
    int d0 = hdr[HDR_MAX0] - hdr[HDR_MIN0] + 1;
    int d1 = hdr[HDR_MAX1] - hdr[HDR_MIN1] + 1;
    int d2 = hdr[HDR_MAX2] - hdr[HDR_MIN2] + 1;
    if (d0 < 1) d0 = 1;
    if (d1 < 1) d1 = 1;
    if (d2 < 1) d2 = 1;
    long long He = (long long)d0 * (long long)d1 * (long long)d2;
    int H = (He > (long long)cap) ? cap : (int)He;
    hdr[HDR_H] = H;
    hdr[HDR_D1] = d1;
    hdr[HDR_D2] = d2;
    hdr[HDR_FLAG] = (H <= SLOTS) ? 1 : 0;   // LDS fast path?
  }
}

__global__ void k_zero_final(float* __restrict__ cnt, float* __restrict__ sx,
                             float* __restrict__ sy, float* __restrict__ sz,
                             const int* __restrict__ hdr) {
  int H = hdr[HDR_H];
  int i = blockIdx.x * blockDim.x + threadIdx.x;
  if (i >= H) return;
  cnt[i] = 0.f; sx[i] = 0.f; sy[i] = 0.f; sz[i] = 0.f;
}

// LDS-aggregated histogram: 128 KB of LDS per workgroup (CDNA5 320KB/WGP).
__global__ __launch_bounds__(HIST_THREADS) void k_hist(
    const float* __restrict__ x, int n, const int* __restrict__ hdr,
    float* __restrict__ cnt, float* __restrict__ sx, float* __restrict__ sy, float* __restrict__ sz,
    float* __restrict__ pcnt, float* __restrict__ psx, float* __restrict__ psy, float* __restrict__ psz) {
  __shared__ unsigned s_c[SLOTS];
  __shared__ float s_x[SLOTS];
  __shared__ float s_y[SLOTS];
  __shared__ float s_z[SLOTS];
  int tid = threadIdx.x;
  int m0 = hdr[HDR_MIN0], m1 = hdr[HDR_MIN1], m2 = hdr[HDR_MIN2];
  int d1 = hdr[HDR_D1], d2 = hdr[HDR_D2];
  int H = hdr[HDR_H];
  int flag = hdr[HDR_FLAG];
  int stride = gridDim.x * blockDim.x;
  int start = blockIdx.x * blockDim.x + tid;
  if (flag) {
    for (int b = tid; b < SLOTS; b += blockDim.x) {
      s_c[b] = 0u; s_x[b] = 0.f; s_y[b] = 0.f; s_z[b] = 0.f;
    }
    __syncthreads();
    for (int i = start; i < n; i += stride) {
      float px = x[3 * i], py = x[3 * i + 1], pz = x[3 * i + 2];
      int v0 = (int)floorf(px * INV_VOXEL) - m0;
      int v1 = (int)floorf(py * INV_VOXEL) - m1;
      int v2 = (int)floorf(pz * INV_VOXEL) - m2;
      int idx = (v0 * d1 + v1) * d2 + v2;
      atomicAdd(&s_c[idx], 1u);
      atomicAdd(&s_x[idx], px);
      atomicAdd(&s_y[idx], py);
      atomicAdd(&s_z[idx], pz);
    }
    __syncthreads();
    size_t base = (size_t)blockIdx.x * SLOTS;   // contention-free slice write
    for (int b = tid; b < SLOTS; b += blockDim.x) {
      pcnt[base + b] = (float)s_c[b];
      psx[base + b] = s_x[b];
      psy[base + b] = s_y[b];
      psz[base + b] = s_z[b];
    }
  } else {
    // fallback for huge grids: direct global atomics
    for (int i = start; i < n; i += stride) {
      float px = x[3 * i], py = x[3 * i + 1], pz = x[3 * i + 2];
      int v0 = (int)floorf(px * INV_VOXEL) - m0;
      int v1 = (int)floorf(py * INV_VOXEL) - m1;
      int v2 = (int)floorf(pz * INV_VOXEL) - m2;
      int idx = (v0 * d1 + v1) * d2 + v2;
      if (idx >= 0 && idx < H) {
        atomicAdd(&cnt[idx], 1.f);
        atomicAdd(&sx[idx], px);
        atomicAdd(&sy[idx], py);
        atomicAdd(&sz[idx], pz);
      }
    }
  }
}

// Reduce NSLICES partial slices per bin via V_WMMA_F32_16X16X4_F32 with
// all-ones A: D[m][n] = sum_k B[k][n]  (tensor-core column sum).
// One wave32 handles 16 bins; B columns are lane&15 (layout-permutation safe).
__global__ __launch_bounds__(256) void k_reduce(
    const int* __restrict__ hdr,
    const float* __restrict__ pcnt, const float* __restrict__ psx,
    const float* __restrict__ psy, const float* __restrict__ psz,
    float* __restrict__ cnt, float* __restrict__ sx, float* __restrict__ sy, float* __restrict__ sz) {
  if (!hdr[HDR_FLAG]) return;
  int H = hdr[HDR_H];
  int wave = threadIdx.x >> 5;
  int lane = threadIdx.x & 31;
  int g = blockIdx.x * 8 + wave;       // 16-bin group per wave
  if (g * 16 >= H) return;             // wave-uniform: EXEC stays all-ones for WMMA
  int half = lane >> 4;
  int bin = g * 16 + (lane & 15);
#ifdef USE_WMMA_REDUCE
  v2f a; a[0] = 1.f; a[1] = 1.f;       // A = 16x4 ones
  v8f cc = {0, 0, 0, 0, 0, 0, 0, 0};
  v8f cx = {0, 0, 0, 0, 0, 0, 0, 0};
  v8f cy = {0, 0, 0, 0, 0, 0, 0, 0};
  v8f cz = {0, 0, 0, 0, 0, 0, 0, 0};
  for (int kk = 0; kk < NSLICES / 4; ++kk) {
    int s0 = kk * 4 + half * 2;
    size_t i0 = (size_t)s0 * SLOTS + bin;
    size_t i1 = i0 + SLOTS;
    v2f bc, bx, by, bz;
    bc[0] = pcnt[i0]; bc[1] = pcnt[i1];
    bx[0] = psx[i0];  bx[1] = psx[i1];
    by[0] = psy[i0];  by[1] = psy[i1];
    bz[0] = psz[i0];  bz[1] = psz[i1];
    cc = __builtin_amdgcn_wmma_f32_16x16x4_f32(false, a, false, bc, (short)0, cc, false, false);
    cx = __builtin_amdgcn_wmma_f32_16x16x4_f32(false, a, false, bx, (short)0, cx, false, false);
    cy = __builtin_amdgcn_wmma_f32_16x16x4_f32(false, a, false, by, (short)0, cy, false, false);
    cz = __builtin_amdgcn_wmma_f32_16x16x4_f32(false, a, false, bz, (short)0, cz, false, false);
  }
  if (half == 0 && bin < H) {
    cnt[bin] = cc[0]; sx[bin] = cx[0]; sy[bin] = cy[0]; sz[bin] = cz[0];
  }
#else
  if (half == 0 && bin < H) {
    float tc = 0.f, tx = 0.f, ty = 0.f, tz = 0.f;
    for (int s = 0; s < NSLICES; ++s) {
      size_t i0 = (size_t)s * SLOTS + bin;
      tc += pcnt[i0]; tx += psx[i0]; ty += psy[i0]; tz += psz[i0];
    }
    cnt[bin] = tc; sx[bin] = tx; sy[bin] = ty; sz[bin] = tz;
  }
#endif
}

// Single-block exclusive scan of occupancy -> rank (sorted-unique row index).
__global__ __launch_bounds__(1024) void k_scan(const float* __restrict__ cnt,
                                               int* __restrict__ rank, int* __restrict__ hdr) {
  __shared__ int s[1024];
  __shared__ int s_run;
  int tid = threadIdx.x;
  if (tid == 0) s_run = 0;
  __syncthreads();
  int H = hdr[HDR_H];
  for (int base = 0; base < H; base += 1024) {
    int i = base + tid;
    int f = (i < H && cnt[i] > 0.f) ? 1 : 0;
    s[tid] = f;
    __syncthreads();
    for (int ofs = 1; ofs < 1024; ofs <<= 1) {
      int t = (tid >= ofs) ? s[tid - ofs] : 0;
      __syncthreads();
      s[tid] += t;
      __syncthreads();
    }
    if (i < H) rank[i] = s_run + s[tid] - f;
    __syncthreads();
    if (tid == 0) s_run += s[1023];
    __syncthreads();
  }
  if (tid == 0) hdr[HDR_NU] = s_run;
}

__global__ void k_zero_out(float* __restrict__ out, int nfloat) {
  int i = blockIdx.x * blockDim.x + threadIdx.x;
  int n4 = nfloat >> 2;
  if (i < n4) ((float4*)out)[i] = make_float4(0.f, 0.f, 0.f, 0.f);
  int tail = nfloat & 3;
  if (i < tail) out[n4 * 4 + i] = 0.f;
}

__global__ void k_scatter(const float* __restrict__ cnt, const float* __restrict__ sx,
                          const float* __restrict__ sy, const float* __restrict__ sz,
                          const int* __restrict__ rank, const int* __restrict__ hdr,
                          float* __restrict__ out) {
  int H = hdr[HDR_H];
  int i = blockIdx.x * blockDim.x + threadIdx.x;
  if (i >= H) return;
  float c = cnt[i];
  if (c > 0.f) {
    int r = rank[i];
    float inv = 1.f / c;
    out[3 * r + 0] = sx[i] * inv;
    out[3 * r + 1] = sy[i] * inv;
    out[3 * r + 2] = sz[i] * inv;
  }
}

extern "C" void kernel_launch(void* const* d_in, const int* in_sizes, int n_in,
                              void* d_out, int out_size, void* d_ws, size_t ws_size,
                              hipStream_t stream) {
  const float* x = (const float*)d_in[0];
  int N = in_sizes[0] / 3;
  float* out = (float*)d_out;

  // Workspace layout
  char* w = (char*)d_ws;
  size_t off = 0;
  int* hdr = (int*)(w + off); off += 256;
  size_t sliceFloats = (size_t)NSLICES * SLOTS;
  float* pcnt = (float*)(w + off); off += sliceFloats * sizeof(float);
  float* psx  = (float*)(w + off); off += sliceFloats * sizeof(float);
  float* psy  = (float*)(w + off); off += sliceFloats * sizeof(float);
  float* psz  = (float*)(w + off); off += sliceFloats * sizeof(float);
  size_t rem = (ws_size > off) ? (ws_size - off) : 0;
  size_t cap = rem / 20;                       // 4 float arrays + 1 int array per bin
  if (cap > (size_t)(1 << 20)) cap = (size_t)(1 << 20);
  if (cap < SLOTS) cap = SLOTS;                // expect ws large enough for fast path
  float* cnt = (float*)(w + off); off += cap * sizeof(float);
  float* sx  = (float*)(w + off); off += cap * sizeof(float);
  float* sy  = (float*)(w + off); off += cap * sizeof(float);
  float* sz  = (float*)(w + off); off += cap * sizeof(float);
  int* rank  = (int*)(w + off);  off += cap * sizeof(int);
  int icap = (int)cap;

  k_init<<<1, 32, 0, stream>>>(hdr);
  k_minmax<<<1024, 256, 0, stream>>>(x, N, hdr);
  k_dims<<<1, 1, 0, stream>>>(hdr, icap);
  int zb = (icap + 255) / 256;
  k_zero_final<<<zb, 256, 0, stream>>>(cnt, sx, sy, sz, hdr);
  k_hist<<<NSLICES, HIST_THREADS, 0, stream>>>(x, N, hdr, cnt, sx, sy, sz, pcnt, psx, psy, psz);
  k_reduce<<<NSLICES, 256, 0, stream>>>(hdr, pcnt, psx, psy, psz, cnt, sx, sy, sz);
  k_scan<<<1, 1024, 0, stream>>>(cnt, rank, hdr);
  int n4 = out_size / 4;
  int zob = (n4 + 255) / 256;
  if (zob < 1) zob = 1;
  k_zero_out<<<zob, 256, 0, stream>>>(out, out_size);
  k_scatter<<<zb, 256, 0, stream>>>(cnt, sx, sy, sz, rank, hdr, out);
}